// MultiBoxLoss_Focal_Giou_10900626997969
// MI455X (gfx1250) — compile-verified
//
#include <hip/hip_runtime.h>
#include <math.h>

// Problem constants (from reference)
#define B_      32
#define P_      24564
#define C_      21
#define O_      100
#define THRESH_ 0.5f
#define VAR0_   0.1f
#define VAR1_   0.2f

#define TPB 256
#define PB  ((P_ + TPB - 1) / TPB)   // 96 tiles of priors

typedef __attribute__((ext_vector_type(2))) float v2f;
typedef __attribute__((ext_vector_type(8))) float v8f;

__device__ __forceinline__ float iou_pt(float ax1, float ay1, float ax2, float ay2,
                                        float bx1, float by1, float bx2, float by2) {
    float ix1 = fmaxf(ax1, bx1), iy1 = fmaxf(ay1, by1);
    float ix2 = fminf(ax2, bx2), iy2 = fminf(ay2, by2);
    float iw = fmaxf(ix2 - ix1, 0.0f), ih = fmaxf(iy2 - iy1, 0.0f);
    float inter = iw * ih;
    float aa = (ax2 - ax1) * (ay2 - ay1);
    float ab = (bx2 - bx1) * (by2 - by1);
    return inter / (aa + ab - inter);
}

// ---------------------------------------------------------------------------
// Kernel 1: per (b, prior) max/argmax IoU over the 100 truths.
// ---------------------------------------------------------------------------
__global__ void k_best_truth(const float* __restrict__ priors,
                             const float* __restrict__ targets,
                             float* __restrict__ bt_ov,
                             int*   __restrict__ bt_idx) {
    __shared__ float tg[O_ * 5];
    int b = blockIdx.x / PB, tile = blockIdx.x % PB;
    for (int i = threadIdx.x; i < O_ * 5; i += TPB) tg[i] = targets[b * O_ * 5 + i];
    __syncthreads();

    int p = tile * TPB + threadIdx.x;
    if (p >= P_) return;

    float cx = priors[p * 4 + 0], cy = priors[p * 4 + 1];
    float w  = priors[p * 4 + 2], h  = priors[p * 4 + 3];
    float px1 = cx - 0.5f * w, py1 = cy - 0.5f * h;
    float px2 = cx + 0.5f * w, py2 = cy + 0.5f * h;

    float best = -1.0f; int bi = 0;
    for (int o = 0; o < O_; ++o) {
        float v = iou_pt(tg[o * 5 + 0], tg[o * 5 + 1], tg[o * 5 + 2], tg[o * 5 + 3],
                         px1, py1, px2, py2);
        if (v > best) { best = v; bi = o; }   // first max wins (jnp.argmax semantics)
    }
    bt_ov [b * P_ + p] = best;
    bt_idx[b * P_ + p] = bi;
}

// ---------------------------------------------------------------------------
// Kernel 2: per (b, truth) argmax IoU over all priors. One block per (b,o).
// ---------------------------------------------------------------------------
__global__ void k_best_prior(const float* __restrict__ priors,
                             const float* __restrict__ targets,
                             int* __restrict__ bp_idx) {
    __shared__ float sv[TPB / 32];
    __shared__ int   si[TPB / 32];

    int b = blockIdx.x / O_, o = blockIdx.x % O_;
    const float* t = &targets[(b * O_ + o) * 5];
    float tx1 = t[0], ty1 = t[1], tx2 = t[2], ty2 = t[3];

    float best = -1.0f; int bi = P_;
    for (int p = threadIdx.x; p < P_; p += TPB) {
        float cx = priors[p * 4 + 0], cy = priors[p * 4 + 1];
        float w  = priors[p * 4 + 2], h  = priors[p * 4 + 3];
        float v = iou_pt(tx1, ty1, tx2, ty2,
                         cx - 0.5f * w, cy - 0.5f * h, cx + 0.5f * w, cy + 0.5f * h);
        if (v > best) { best = v; bi = p; }
    }
    // wave32 argmax reduce, lowest-index tie-break
    for (int off = 16; off > 0; off >>= 1) {
        float ov = __shfl_xor(best, off, 32);
        int   oi = __shfl_xor(bi,   off, 32);
        if (ov > best || (ov == best && oi < bi)) { best = ov; bi = oi; }
    }
    int wv = threadIdx.x >> 5;
    if ((threadIdx.x & 31) == 0) { sv[wv] = best; si[wv] = bi; }
    __syncthreads();
    if (threadIdx.x == 0) {
        for (int i = 1; i < TPB / 32; ++i)
            if (sv[i] > best || (sv[i] == best && si[i] < bi)) { best = sv[i]; bi = si[i]; }
        bp_idx[b * O_ + o] = bi;
    }
}

// ---------------------------------------------------------------------------
// Kernel 3: apply the scatter overrides serially per batch (last write wins,
// matching JAX .at[].set with duplicate indices) + zero the accumulators.
// ---------------------------------------------------------------------------
__global__ void k_scatter(const int* __restrict__ bp_idx,
                          float* __restrict__ bt_ov,
                          int*   __restrict__ bt_idx,
                          float* __restrict__ acc) {
    int b = threadIdx.x;
    if (b == 0) { acc[0] = 0.0f; acc[1] = 0.0f; acc[2] = 0.0f; }
    if (b < B_) {
        for (int o = 0; o < O_; ++o) {
            int p = bp_idx[b * O_ + o];
            bt_ov [b * P_ + p] = 2.0f;
            bt_idx[b * P_ + p] = o;
        }
    }
}

// ---------------------------------------------------------------------------
// Kernel 4: GIoU loc loss + focal conf loss; WMMA block reduction.
// ---------------------------------------------------------------------------
__global__ void k_loss(const float* __restrict__ loc,
                       const float* __restrict__ conf,
                       const float* __restrict__ priors,
                       const float* __restrict__ targets,
                       const float* __restrict__ bt_ov,
                       const int*   __restrict__ bt_idx,
                       float* __restrict__ acc) {
    __shared__ float tg[O_ * 5];
    __shared__ float red[3][TPB];

    int b = blockIdx.x / PB, tile = blockIdx.x % PB;
    int p = tile * TPB + threadIdx.x;

    for (int i = threadIdx.x; i < O_ * 5; i += TPB) tg[i] = targets[b * O_ * 5 + i];
    __syncthreads();

    float tl = 0.0f, tc = 0.0f, tn = 0.0f;
    if (p < P_) {
        const float* crow = &conf[((size_t)(b * P_ + p)) * C_];
        __builtin_prefetch(crow, 0, 1);            // global_prefetch_b8: hide conf latency

        int   idx = bt_idx[b * P_ + p];
        float ov  = bt_ov [b * P_ + p];
        int cls = (ov < THRESH_) ? 0 : ((int)tg[idx * 5 + 4] + 1);
        bool pos = cls > 0;

        // decode predicted box
        float cx = priors[p * 4 + 0], cy = priors[p * 4 + 1];
        float pw = priors[p * 4 + 2], ph = priors[p * 4 + 3];
        size_t lbase = ((size_t)(b * P_ + p)) * 4;
        float l0 = loc[lbase + 0], l1 = loc[lbase + 1];
        float l2 = loc[lbase + 2], l3 = loc[lbase + 3];
        float dcx = cx + l0 * VAR0_ * pw;
        float dcy = cy + l1 * VAR0_ * ph;
        float dw  = pw * __expf(l2 * VAR1_);
        float dh  = ph * __expf(l3 * VAR1_);
        float dx1 = dcx - 0.5f * dw, dy1 = dcy - 0.5f * dh;
        float dx2 = dcx + 0.5f * dw, dy2 = dcy + 0.5f * dh;

        // matched truth box
        float mx1 = tg[idx * 5 + 0], my1 = tg[idx * 5 + 1];
        float mx2 = tg[idx * 5 + 2], my2 = tg[idx * 5 + 3];

        // GIoU
        float a1 = (dx2 - dx1) * (dy2 - dy1);
        float a2 = (mx2 - mx1) * (my2 - my1);
        float ix1 = fmaxf(dx1, mx1), iy1 = fmaxf(dy1, my1);
        float ix2 = fminf(dx2, mx2), iy2 = fminf(dy2, my2);
        float inter = fmaxf(ix2 - ix1, 0.0f) * fmaxf(iy2 - iy1, 0.0f);
        float uni = a1 + a2 - inter;
        float ex1 = fminf(dx1, mx1), ey1 = fminf(dy1, my1);
        float ex2 = fmaxf(dx2, mx2), ey2 = fmaxf(dy2, my2);
        float enc = fmaxf(ex2 - ex1, 0.0f) * fmaxf(ey2 - ey1, 0.0f);
        float giou = inter / uni - (enc - uni) / enc;
        if (pos) { tl = 5.0f * (1.0f - giou); tn = 1.0f; }

        // focal-modulated softmax CE over C_=21 logits (register resident)
        float x[C_];
        float m = -3.402823466e38f;
#pragma unroll
        for (int c = 0; c < C_; ++c) { x[c] = crow[c]; m = fmaxf(m, x[c]); }
        float s = 0.0f;
#pragma unroll
        for (int c = 0; c < C_; ++c) s += __expf(x[c] - m);
        float log_pt = crow[cls] - m - __logf(s);   // dynamic index via cached re-read
        float pt = __expf(log_pt);
        float om = 1.0f - pt;
        tc = -om * om * log_pt;
    }

    red[0][threadIdx.x] = tl;
    red[1][threadIdx.x] = tc;
    red[2][threadIdx.x] = tn;
    __syncthreads();

    // Block reduction on the matrix pipe: D = ones(16x4) x B(4x16) + C gives
    // column sums of B; 4 chained WMMAs fold all 256 partials per quantity.
    if (threadIdx.x < 32) {
        int lane = threadIdx.x;
        v2f aone; aone.x = 1.0f; aone.y = 1.0f;
        float sums[3];
#pragma unroll
        for (int q = 0; q < 3; ++q) {
            v8f cacc = {0.f, 0.f, 0.f, 0.f, 0.f, 0.f, 0.f, 0.f};
#pragma unroll
            for (int ch = 0; ch < 4; ++ch) {
                v2f bv;
                bv.x = red[q][ch * 64 + lane * 2 + 0];
                bv.y = red[q][ch * 64 + lane * 2 + 1];
                cacc = __builtin_amdgcn_wmma_f32_16x16x4_f32(
                    false, aone, false, bv, (short)0, cacc, false, false);
            }
            // every row of D holds the 16 column sums -> c[0] over 32 lanes = 2x total
            float v = cacc[0];
            for (int off = 16; off > 0; off >>= 1) v += __shfl_xor(v, off, 32);
            sums[q] = v * 0.5f;
        }
        if (lane == 0) {
            atomicAdd(&acc[0], sums[0]);
            atomicAdd(&acc[1], sums[1]);
            atomicAdd(&acc[2], sums[2]);
        }
    }
}

// ---------------------------------------------------------------------------
// Kernel 5: finalize the two scalars.
// ---------------------------------------------------------------------------
__global__ void k_final(const float* __restrict__ acc, float* __restrict__ out) {
    if (threadIdx.x == 0) {
        float N = acc[2];
        out[0] = acc[0] / N;
        out[1] = acc[1] / N;
    }
}

extern "C" void kernel_launch(void* const* d_in, const int* in_sizes, int n_in,
                              void* d_out, int out_size, void* d_ws, size_t ws_size,
                              hipStream_t stream) {
    const float* loc     = (const float*)d_in[0];   // (B,P,4)
    const float* conf    = (const float*)d_in[1];   // (B,P,C)
    const float* priors  = (const float*)d_in[2];   // (P,4)
    const float* targets = (const float*)d_in[3];   // (B,O,5)

    char* ws = (char*)d_ws;
    size_t off = 0;
    float* bt_ov  = (float*)(ws + off); off += (size_t)B_ * P_ * sizeof(float);
    int*   bt_idx = (int*)  (ws + off); off += (size_t)B_ * P_ * sizeof(int);
    int*   bp_idx = (int*)  (ws + off); off += (size_t)B_ * O_ * sizeof(int);
    float* acc    = (float*)(ws + off); off += 4 * sizeof(float);
    float* out    = (float*)d_out;
    (void)in_sizes; (void)n_in; (void)out_size; (void)ws_size;

    k_best_truth<<<B_ * PB, TPB, 0, stream>>>(priors, targets, bt_ov, bt_idx);
    k_best_prior<<<B_ * O_, TPB, 0, stream>>>(priors, targets, bp_idx);
    k_scatter  <<<1, 32, 0, stream>>>(bp_idx, bt_ov, bt_idx, acc);
    k_loss     <<<B_ * PB, TPB, 0, stream>>>(loc, conf, priors, targets, bt_ov, bt_idx, acc);
    k_final    <<<1, 1, 0, stream>>>(acc, out);
}